// LocalFeatureAggregation_8787503088234
// MI455X (gfx1250) — compile-verified
//
#include <hip/hip_runtime.h>

#define NPTS 16384
#define NBATCH 2
#define KNN 16
#define SPLIT 4                 // candidate-range splits for kNN occupancy
#define NTILES (NPTS / 16)      // 1024 point tiles
#define TILES_PER_SPLIT (NTILES / SPLIT)

typedef __attribute__((ext_vector_type(2))) float v2f;
typedef __attribute__((ext_vector_type(8))) float v8f;

__device__ __forceinline__ v8f wmma_f32_16x16x4(v2f a, v2f b, v8f c) {
  // D = A(16x4,f32) * B(4x16,f32) + C(16x16,f32)
  return __builtin_amdgcn_wmma_f32_16x16x4_f32(false, a, false, b, (short)0, c,
                                               false, false);
}

// Branchless sorted-ascending top-16 insertion (cmp + cndmask straight line).
__device__ __forceinline__ void topk_insert(float (&kd)[KNN], int (&ki)[KNN],
                                            float dd, int ii) {
#pragma unroll
  for (int s = 0; s < KNN; ++s) {
    const float ok = kd[s];
    const int oi = ki[s];
    const bool sw = dd < ok;
    kd[s] = sw ? dd : ok;
    ki[s] = sw ? ii : oi;
    dd = sw ? ok : dd;
    ii = sw ? oi : ii;
  }
}

// ---------------------------------------------------------------------------
// Kernel 1a: partial brute-force kNN via V_WMMA_F32_16X16X4_F32.
// One wave = 32 queries x 1/SPLIT of the candidate range (256 tiles of 16).
// 4096 waves total -> enough occupancy to hide the LDS round-trip latency.
// Ranks (-2 q.p + |p|^2); |q|^2 added at merge (per-row constant, order-safe).
// ---------------------------------------------------------------------------
__global__ __launch_bounds__(256) void knn_partial_kernel(
    const float* __restrict__ coords, int* __restrict__ pidx,
    float* __restrict__ pdist) {
  __shared__ float ldist[8][32][17];  // per-wave 32x16 tile, padded stride
  const int lane = threadIdx.x & 31;
  const int wv = threadIdx.x >> 5;
  const int gw = blockIdx.x * 8 + wv;          // 4096 waves total
  const int bqw = gw & 1023;                   // (batch, query-group)
  const int split = gw >> 10;                  // candidate-range split
  const int b = bqw >> 9;                      // 512 query-waves per batch
  const int qbase = (bqw & 511) << 5;          // 32 queries per wave
  const int lh = lane & 15;
  const bool hi = lane >= 16;

  // A fragments (two 16-row query tiles), scaled by -2.
  // f32 16x4 A layout: lanes 0-15 hold K=0(x),K=1(y); lanes 16-31 hold K=2(z),K=3(0)
  const float* q0 = coords + ((size_t)b * NPTS + qbase + lh) * 3;
  const float* q1 = q0 + 16 * 3;
  v2f a0, a1;
  {
    float x0 = q0[0], y0 = q0[1], z0 = q0[2];
    float x1 = q1[0], y1 = q1[1], z1 = q1[2];
    a0.x = -2.0f * (hi ? z0 : x0);
    a0.y = hi ? 0.0f : -2.0f * y0;
    a1.x = -2.0f * (hi ? z1 : x1);
    a1.y = hi ? 0.0f : -2.0f * y1;
  }

  float kd[KNN];
  int ki[KNN];
#pragma unroll
  for (int s = 0; s < KNN; ++s) { kd[s] = 3.4e38f; ki[s] = 0; }

  const int j0 = split * TILES_PER_SPLIT;
  const int j1 = j0 + TILES_PER_SPLIT;
  for (int j = j0; j < j1; ++j) {
    const float* pp = coords + ((size_t)b * NPTS + j * 16 + lh) * 3;
    float px = pp[0], py = pp[1], pz = pp[2];
    if (j + 1 < j1) __builtin_prefetch(pp + 48, 0, 3);  // next point tile
    float sqp = px * px + py * py + pz * pz;

    // B fragment: 4x16, K layout mirrors A: lanes0-15 -> K=0,1 ; lanes16-31 -> K=2,3
    v2f bf;
    bf.x = hi ? pz : px;
    bf.y = hi ? 0.0f : py;

    v8f cz = {};
    v8f d0 = wmma_f32_16x16x4(a0, bf, cz);
    v8f d1 = wmma_f32_16x16x4(a1, bf, cz);

    // D layout: VGPR v -> row (hi ? v+8 : v), col = lane%16
    const int rb = hi ? 8 : 0;
#pragma unroll
    for (int v = 0; v < 8; ++v) {
      ldist[wv][rb + v][lh] = d0[v] + sqp;
      ldist[wv][16 + rb + v][lh] = d1[v] + sqp;
    }
    __syncthreads();

    // bulk-load this lane's query row, then rare-case guarded inserts
    float cand[16];
#pragma unroll
    for (int cc = 0; cc < 16; ++cc) cand[cc] = ldist[wv][lane][cc];
#pragma unroll
    for (int cc = 0; cc < 16; ++cc) {
      if (cand[cc] < kd[KNN - 1]) topk_insert(kd, ki, cand[cc], j * 16 + cc);
    }
    __syncthreads();
  }

  const size_t q = (size_t)b * NPTS + qbase + lane;   // global query id
  float* od = pdist + (q * SPLIT + split) * KNN;
  int* oi = pidx + (q * SPLIT + split) * KNN;
#pragma unroll
  for (int s = 0; s < KNN; ++s) {
    od[s] = kd[s];
    oi[s] = ki[s];
  }
}

// ---------------------------------------------------------------------------
// Kernel 1b: merge SPLIT sorted partial lists per query -> final top-16.
// Lists are sorted ascending, so each list stops early once it can't qualify.
// Adds the per-query |q|^2 so stored distances match the reference.
// ---------------------------------------------------------------------------
__global__ __launch_bounds__(256) void knn_merge_kernel(
    const float* __restrict__ coords, const int* __restrict__ pidx,
    const float* __restrict__ pdist, int* __restrict__ fidx,
    float* __restrict__ fdist) {
  const size_t q = (size_t)blockIdx.x * 256 + threadIdx.x;  // B*N threads
  const float* cs = coords + q * 3;
  const float sqq = cs[0] * cs[0] + cs[1] * cs[1] + cs[2] * cs[2];

  float kd[KNN];
  int ki[KNN];
#pragma unroll
  for (int s = 0; s < KNN; ++s) { kd[s] = 3.4e38f; ki[s] = 0; }

  for (int sp = 0; sp < SPLIT; ++sp) {
    const float* pd = pdist + (q * SPLIT + sp) * KNN;
    const int* pi = pidx + (q * SPLIT + sp) * KNN;
    for (int s = 0; s < KNN; ++s) {
      const float dc = pd[s];
      if (dc >= kd[KNN - 1]) break;  // sorted: rest can't qualify
      topk_insert(kd, ki, dc, pi[s]);
    }
  }
#pragma unroll
  for (int s = 0; s < KNN; ++s) {
    fdist[q * KNN + s] = kd[s] + sqq;
    fidx[q * KNN + s] = ki[s];
  }
}

// ---------------------------------------------------------------------------
// Kernel 2: x1 = leaky_relu(mlp1_w(16x8) @ features + b, 0.2)   (B,16,N)
// ---------------------------------------------------------------------------
__global__ __launch_bounds__(256) void mlp1_kernel(const float* __restrict__ feats,
                                                   const float* __restrict__ w,
                                                   const float* __restrict__ bias,
                                                   float* __restrict__ x1) {
  const int t = blockIdx.x * 256 + threadIdx.x;  // B*N threads
  const int b = t / NPTS, n = t % NPTS;
  float f[8];
#pragma unroll
  for (int ci = 0; ci < 8; ++ci) f[ci] = feats[((size_t)b * 8 + ci) * NPTS + n];
#pragma unroll
  for (int c = 0; c < 16; ++c) {
    float acc = bias[c];
#pragma unroll
    for (int ci = 0; ci < 8; ++ci) acc += w[c * 8 + ci] * f[ci];
    x1[((size_t)b * 16 + c) * NPTS + n] = acc > 0.f ? acc : 0.2f * acc;
  }
}

// ---------------------------------------------------------------------------
// Kernel 3: fused LocalSpatialEncoding + AttentivePooling.
// 16 threads per point (one per neighbor k); softmax over the 16-lane k-group
// via wave shuffles; lane k==0 writes the pooled CO-channel output.
// fin: (B,16,N), out: (B,CO,N)
// ---------------------------------------------------------------------------
__global__ __launch_bounds__(256) void lse_pool_kernel(
    const float* __restrict__ coords, const float* __restrict__ fin,
    const int* __restrict__ kidx, const float* __restrict__ kdist,
    const float* __restrict__ lw, const float* __restrict__ lb,
    const float* __restrict__ lg, const float* __restrict__ lbeta,
    const float* __restrict__ sw, const float* __restrict__ pw,
    const float* __restrict__ pb, const float* __restrict__ pg,
    const float* __restrict__ pbeta, float* __restrict__ out, int CO) {
  const int k = threadIdx.x & 15;
  const int p = threadIdx.x >> 4;
  const int pt = blockIdx.x * 16 + p;
  const int b = pt / NPTS;
  const int n = pt % NPTS;
  const float bns = 1.0f / sqrtf(1.0f + 1e-6f);

  const float* cen = coords + ((size_t)b * NPTS + n) * 3;
  const float cx = cen[0], cy = cen[1], cz = cen[2];
  const size_t kb = ((size_t)b * NPTS + n) * KNN + k;
  const int j = kidx[kb];
  const float dd = kdist[kb];
  const float* nb = coords + ((size_t)b * NPTS + j) * 3;
  const float nx = nb[0], ny = nb[1], nz = nb[2];
  const float rel[10] = {cx, cy, cz, nx, ny, nz, cx - nx, cy - ny, cz - nz, dd};

  float x2[32];
#pragma unroll
  for (int c = 0; c < 16; ++c) {
    float acc = lb[c];
#pragma unroll
    for (int i = 0; i < 10; ++i) acc += lw[c * 10 + i] * rel[i];
    acc = acc * (lg[c] * bns) + lbeta[c];
    x2[c] = acc > 0.f ? acc : 0.f;
  }
#pragma unroll
  for (int c = 0; c < 16; ++c)
    x2[16 + c] = fin[((size_t)b * 16 + c) * NPTS + n];

  float pooled[32];
#pragma unroll
  for (int o = 0; o < 32; ++o) {
    float s = 0.f;
#pragma unroll
    for (int i = 0; i < 32; ++i) s += sw[o * 32 + i] * x2[i];
    float m = s;
#pragma unroll
    for (int off = 8; off > 0; off >>= 1) m = fmaxf(m, __shfl_xor(m, off, 16));
    const float e = __expf(s - m);
    float num = e * x2[o], den = e;
#pragma unroll
    for (int off = 8; off > 0; off >>= 1) {
      num += __shfl_xor(num, off, 16);
      den += __shfl_xor(den, off, 16);
    }
    pooled[o] = num / den;
  }

  if (k == 0) {
    for (int co = 0; co < CO; ++co) {
      float acc = pb[co];
#pragma unroll
      for (int i = 0; i < 32; ++i) acc += pw[co * 32 + i] * pooled[i];
      acc = acc * (pg[co] * bns) + pbeta[co];
      out[((size_t)b * CO + co) * NPTS + n] = acc > 0.f ? acc : 0.f;
    }
  }
}

// ---------------------------------------------------------------------------
// Kernel 4: out = leaky(mlp2_w(64x32) @ x5 + b + BN(sc_w(64x8) @ feats + scb), 0.01)
// WMMA GEMM: wave handles 16 points x 64 out-channels (4 row tiles, K=32 in 8
// chained wmma_f32_16x16x4 steps). C is pre-seeded with bias + shortcut.
// ---------------------------------------------------------------------------
__global__ __launch_bounds__(256) void final_kernel(
    const float* __restrict__ x5, const float* __restrict__ feats,
    const float* __restrict__ mw, const float* __restrict__ mb,
    const float* __restrict__ scw, const float* __restrict__ scb,
    const float* __restrict__ scg, const float* __restrict__ scbeta,
    float* __restrict__ out) {
  const int lane = threadIdx.x & 31;
  const int wv = threadIdx.x >> 5;
  const int gw = blockIdx.x * 8 + wv;       // 2048 waves = B*N/16
  const int b = gw / (NPTS / 16);
  const int nbase = (gw % (NPTS / 16)) * 16;
  const int lh = lane & 15;
  const bool hi = lane >= 16;
  const int n = nbase + lh;                 // this lane's column
  const float bns = 1.0f / sqrtf(1.0f + 1e-6f);

  // B fragments for 8 K-steps: channel = kk*4 + (hi?2:0) + v, col = n
  v2f bf[8];
#pragma unroll
  for (int kk = 0; kk < 8; ++kk) {
    const int ch0 = kk * 4 + (hi ? 2 : 0);
    bf[kk].x = x5[((size_t)b * 32 + ch0) * NPTS + n];
    bf[kk].y = x5[((size_t)b * 32 + ch0 + 1) * NPTS + n];
  }

  float f[8];
#pragma unroll
  for (int ci = 0; ci < 8; ++ci) f[ci] = feats[((size_t)b * 8 + ci) * NPTS + n];

#pragma unroll
  for (int t = 0; t < 4; ++t) {
    // seed C with mlp2 bias + BN'd shortcut for (row co, col n)
    v8f c;
#pragma unroll
    for (int v = 0; v < 8; ++v) {
      const int co = t * 16 + (hi ? v + 8 : v);
      float acc = scb[co];
#pragma unroll
      for (int ci = 0; ci < 8; ++ci) acc += scw[co * 8 + ci] * f[ci];
      acc = acc * (scg[co] * bns) + scbeta[co];
      c[v] = acc + mb[co];
    }
    // A fragments: row = t*16 + lane%16, K slice per lane-half
#pragma unroll
    for (int kk = 0; kk < 8; ++kk) {
      const int row = t * 16 + lh;
      const int ch0 = kk * 4 + (hi ? 2 : 0);
      v2f af;
      af.x = mw[row * 32 + ch0];
      af.y = mw[row * 32 + ch0 + 1];
      c = wmma_f32_16x16x4(af, bf[kk], c);
    }
#pragma unroll
    for (int v = 0; v < 8; ++v) {
      const int co = t * 16 + (hi ? v + 8 : v);
      const float y = c[v];
      out[((size_t)b * 64 + co) * NPTS + n] = y > 0.f ? y : 0.01f * y;
    }
  }
}

// ---------------------------------------------------------------------------
extern "C" void kernel_launch(void* const* d_in, const int* in_sizes, int n_in,
                              void* d_out, int out_size, void* d_ws, size_t ws_size,
                              hipStream_t stream) {
  const float* coords     = (const float*)d_in[0];
  const float* features   = (const float*)d_in[1];
  const float* mlp1_w     = (const float*)d_in[2];
  const float* mlp1_b     = (const float*)d_in[3];
  const float* lse1_w     = (const float*)d_in[4];
  const float* lse1_b     = (const float*)d_in[5];
  const float* lse1_g     = (const float*)d_in[6];
  const float* lse1_beta  = (const float*)d_in[7];
  const float* p1_score_w = (const float*)d_in[8];
  const float* p1_w       = (const float*)d_in[9];
  const float* p1_b       = (const float*)d_in[10];
  const float* p1_g       = (const float*)d_in[11];
  const float* p1_beta    = (const float*)d_in[12];
  const float* lse2_w     = (const float*)d_in[13];
  const float* lse2_b     = (const float*)d_in[14];
  const float* lse2_g     = (const float*)d_in[15];
  const float* lse2_beta  = (const float*)d_in[16];
  const float* p2_score_w = (const float*)d_in[17];
  const float* p2_w       = (const float*)d_in[18];
  const float* p2_b       = (const float*)d_in[19];
  const float* p2_g       = (const float*)d_in[20];
  const float* p2_beta    = (const float*)d_in[21];
  const float* mlp2_w     = (const float*)d_in[22];
  const float* mlp2_b     = (const float*)d_in[23];
  const float* sc_w       = (const float*)d_in[24];
  const float* sc_b       = (const float*)d_in[25];
  const float* sc_g       = (const float*)d_in[26];
  const float* sc_beta    = (const float*)d_in[27];

  // Workspace layout (20 MB) with region reuse:
  //   [0 .. 16MB)  partial kNN lists (dead after merge) -> later x1/x3/x5
  //   [16 .. 20MB) final kNN idx/dist (live through both lse_pool calls)
  char* ws = (char*)d_ws;
  float* pdist = (float*)(ws);                             // 8 MB
  int*   pidx  = (int*)(ws + (size_t)(8u << 20));          // 8 MB
  int*   fidx  = (int*)(ws + (size_t)(16u << 20));         // 2 MB
  float* fdist = (float*)(ws + (size_t)(18u << 20));       // 2 MB
  float* x1    = (float*)(ws);                             // 2 MB (B,16,N)
  float* x3    = (float*)(ws + (size_t)(2u << 20));        // 2 MB (B,16,N)
  float* x5    = (float*)(ws + (size_t)(4u << 20));        // 4 MB (B,32,N)

  knn_partial_kernel<<<NBATCH * (NPTS / 32) * SPLIT / 8, 256, 0, stream>>>(
      coords, pidx, pdist);
  knn_merge_kernel<<<NBATCH * NPTS / 256, 256, 0, stream>>>(
      coords, pidx, pdist, fidx, fdist);
  mlp1_kernel<<<NBATCH * NPTS / 256, 256, 0, stream>>>(features, mlp1_w, mlp1_b, x1);
  lse_pool_kernel<<<NBATCH * NPTS / 16, 256, 0, stream>>>(
      coords, x1, fidx, fdist, lse1_w, lse1_b, lse1_g, lse1_beta,
      p1_score_w, p1_w, p1_b, p1_g, p1_beta, x3, 16);
  lse_pool_kernel<<<NBATCH * NPTS / 16, 256, 0, stream>>>(
      coords, x3, fidx, fdist, lse2_w, lse2_b, lse2_g, lse2_beta,
      p2_score_w, p2_w, p2_b, p2_g, p2_beta, x5, 32);
  final_kernel<<<NBATCH * NPTS / 16 / 8, 256, 0, stream>>>(
      x5, features, mlp2_w, mlp2_b, sc_w, sc_b, sc_g, sc_beta, (float*)d_out);

  (void)in_sizes; (void)n_in; (void)out_size; (void)ws_size;
}